// StandGCNX_15839839387791
// MI455X (gfx1250) — compile-verified
//
#include <hip/hip_runtime.h>

// ---------------------------------------------------------------------------
// StandGCNX (3-layer GCN) for MI455X / gfx1250, wave32.
//   layer: h = x @ W ; out = D^-1/2 (A+I) D^-1/2 h + b ; relu (layers 1,2)
// GEMM uses native fp32 WMMA (v_wmma_f32_16x16x4_f32) -> full fp32 precision.
// Scatter uses global_atomic_add_f32 (unsafeAtomicAdd); graph fits in 192MB L2.
// ---------------------------------------------------------------------------

typedef __attribute__((ext_vector_type(2))) float v2f;
typedef __attribute__((ext_vector_type(8))) float v8f;

#define N_NODES 100000
#define N_EDGES 1600000
#define NFEAT   128
#define NHID    128
#define NCLASS  40

// ---------------------------------------------------------------------------
// Dense GEMM C[M,NOUT] = A[M,K] * W[K,NOUT] via V_WMMA_F32_16X16X4_F32.
// Block = 256 threads = 8 waves. Wave w computes rows [16*(8*bid+w), +16)
// across all NOUT columns (NTILES tiles of 16). W staged in LDS (padded to
// NTILES*16 cols). K looped in steps of 4 (fp32 WMMA K).
//
// Fragment layouts (ISA 7.12.2, wave32):
//   A 16x4 : lanes 0-15 -> M=lane, v[0]=K0 v[1]=K1 ; lanes 16-31 -> K2,K3
//   B 4x16 : lanes 0-15 -> N=lane, v[0]=K0 v[1]=K1 ; lanes 16-31 -> K2,K3
//   C 16x16: VGPR r, lanes 0-15 -> row r ; lanes 16-31 -> row r+8
// ---------------------------------------------------------------------------
template <int K, int NOUT, int NTILES>
__global__ __launch_bounds__(256)
void gemm_wmma_f32(const float* __restrict__ A, const float* __restrict__ W,
                   float* __restrict__ C, int M) {
  constexpr int NP = NTILES * 16;
  __shared__ float sW[K][NP];

  const int tid = threadIdx.x;
  for (int idx = tid; idx < K * NP; idx += 256) {
    const int r = idx / NP, c = idx % NP;
    sW[r][c] = (c < NOUT) ? W[r * NOUT + c] : 0.0f;
  }
  __syncthreads();

  const int wave = tid >> 5;
  const int lane = tid & 31;
  const int tileRow = blockIdx.x * 8 + wave;       // wave-uniform
  if (tileRow * 16 >= M) return;                   // wave-uniform exit: EXEC stays all-1s
  const int m0 = tileRow * 16;
  const int hi = lane >> 4;                        // 0: lanes 0-15, 1: lanes 16-31
  const int ln = lane & 15;

  const float* __restrict__ arow = A + (size_t)(m0 + ln) * K;

  v8f acc[NTILES];
#pragma unroll
  for (int t = 0; t < NTILES; ++t) acc[t] = (v8f)0.0f;

  for (int k = 0; k < K; k += 4) {
    v2f afrag;
    afrag.x = arow[k + 2 * hi];                    // merges to global_load_b64
    afrag.y = arow[k + 2 * hi + 1];
#pragma unroll
    for (int t = 0; t < NTILES; ++t) {
      v2f bfrag;
      bfrag.x = sW[k + 2 * hi][t * 16 + ln];
      bfrag.y = sW[k + 2 * hi + 1][t * 16 + ln];
      acc[t] = __builtin_amdgcn_wmma_f32_16x16x4_f32(
          /*neg_a=*/false, afrag, /*neg_b=*/false, bfrag,
          /*c_mod=*/(short)0, acc[t], /*reuse_a=*/false, /*reuse_b=*/false);
    }
  }

#pragma unroll
  for (int t = 0; t < NTILES; ++t) {
    const int col = t * 16 + ln;
    if (col < NOUT) {
#pragma unroll
      for (int r = 0; r < 8; ++r) {
        const int m = m0 + r + hi * 8;
        C[(size_t)m * NOUT + col] = acc[t][r];
      }
    }
  }
}

// ---------------------------------------------------------------------------
// Graph-normalization helpers
// ---------------------------------------------------------------------------
__global__ __launch_bounds__(256) void zero_f32(float* __restrict__ p, int n) {
  const int i = blockIdx.x * 256 + threadIdx.x;
  if (i < n) p[i] = 0.0f;
}

__global__ __launch_bounds__(256) void deg_count(const long long* __restrict__ dst,
                                                 float* __restrict__ deg, int e) {
  const int i = blockIdx.x * 256 + threadIdx.x;
  if (i < e) unsafeAtomicAdd(&deg[dst[i]], 1.0f);   // global_atomic_add_f32
}

__global__ __launch_bounds__(256) void deg_to_dinv(float* __restrict__ deg, int n) {
  const int i = blockIdx.x * 256 + threadIdx.x;
  if (i < n) {
    const float d = deg[i] + 1.0f;                  // +1 self-loop; always > 0
    deg[i] = rsqrtf(d);
  }
}

// out[i,f] = dinv[i]^2 * h[i,f]   (self-loop term; initializes accumulator)
template <int F>
__global__ __launch_bounds__(256)
void self_loop_init(const float* __restrict__ h, const float* __restrict__ dinv,
                    float* __restrict__ out, int n) {
  const int i = blockIdx.x * 256 + threadIdx.x;
  if (i < n * F) {
    const int node = i / F;
    const float di = dinv[node];
    out[i] = di * di * h[i];
  }
}

// One wave per edge, float4 per lane (32 lanes * 4 = 128 features).
__global__ __launch_bounds__(256)
void scatter128(const float* __restrict__ h, const long long* __restrict__ src,
                const long long* __restrict__ dst, const float* __restrict__ dinv,
                float* __restrict__ out, int e) {
  const int tid = blockIdx.x * 256 + threadIdx.x;
  const int edge = tid >> 5;
  if (edge >= e) return;
  const int j = tid & 31;
  const long long s = src[edge];                    // uniform within wave
  const long long d = dst[edge];
  const float norm = dinv[s] * dinv[d];
  const float4 v = *(const float4*)(h + (size_t)s * 128 + j * 4);
  float* o = out + (size_t)d * 128 + j * 4;
  unsafeAtomicAdd(o + 0, v.x * norm);
  unsafeAtomicAdd(o + 1, v.y * norm);
  unsafeAtomicAdd(o + 2, v.z * norm);
  unsafeAtomicAdd(o + 3, v.w * norm);
}

// Generic feature count (used for NCLASS=40): one wave per edge, strided lanes.
template <int F>
__global__ __launch_bounds__(256)
void scatterF(const float* __restrict__ h, const long long* __restrict__ src,
              const long long* __restrict__ dst, const float* __restrict__ dinv,
              float* __restrict__ out, int e) {
  const int tid = blockIdx.x * 256 + threadIdx.x;
  const int edge = tid >> 5;
  if (edge >= e) return;
  const int j = tid & 31;
  const long long s = src[edge];
  const long long d = dst[edge];
  const float norm = dinv[s] * dinv[d];
  for (int f = j; f < F; f += 32)
    unsafeAtomicAdd(out + (size_t)d * F + f, h[(size_t)s * F + f] * norm);
}

template <int F, bool RELU>
__global__ __launch_bounds__(256)
void bias_act(float* __restrict__ out, const float* __restrict__ b, int n) {
  const int i = blockIdx.x * 256 + threadIdx.x;
  if (i < n * F) {
    const float v = out[i] + b[i % F];
    out[i] = RELU ? fmaxf(v, 0.0f) : v;
  }
}

// ---------------------------------------------------------------------------
// Host-side orchestration (all on `stream`, graph-capture safe)
// ---------------------------------------------------------------------------
extern "C" void kernel_launch(void* const* d_in, const int* in_sizes, int n_in,
                              void* d_out, int out_size, void* d_ws, size_t ws_size,
                              hipStream_t stream) {
  (void)in_sizes; (void)n_in; (void)out_size; (void)ws_size;

  const float*      x   = (const float*)d_in[0];
  const long long*  ei  = (const long long*)d_in[1];   // [2, E] int64
  const long long*  src = ei;
  const long long*  dst = ei + N_EDGES;
  const float* W1 = (const float*)d_in[2];
  const float* b1 = (const float*)d_in[3];
  const float* Wm = (const float*)d_in[4];
  const float* bm = (const float*)d_in[5];
  const float* W2 = (const float*)d_in[6];
  const float* b2 = (const float*)d_in[7];
  float* out = (float*)d_out;                          // [N_NODES, NCLASS] fp32

  // Workspace layout: dinv | bufA (N*128) | bufB (N*128)
  float* dinv = (float*)d_ws;
  float* bufA = dinv + N_NODES;
  float* bufB = bufA + (size_t)N_NODES * NHID;

  const int nThreads = 256;
  const int nodeBlk  = (N_NODES + nThreads - 1) / nThreads;
  const int edgeBlk  = (N_EDGES + nThreads - 1) / nThreads;
  const int nf128Blk = ((N_NODES * 128) + nThreads - 1) / nThreads;
  const int nf40Blk  = ((N_NODES * NCLASS) + nThreads - 1) / nThreads;
  const int edgeWaveBlk = ((N_EDGES * 32) + nThreads - 1) / nThreads; // wave/edge
  const int gemmBlk  = ((N_NODES / 16) + 7) / 8;       // 8 row-tiles per block

  // --- normalization: deg -> rsqrt(deg+1), computed once, reused all layers
  zero_f32<<<nodeBlk, nThreads, 0, stream>>>(dinv, N_NODES);
  deg_count<<<edgeBlk, nThreads, 0, stream>>>(dst, dinv, N_EDGES);
  deg_to_dinv<<<nodeBlk, nThreads, 0, stream>>>(dinv, N_NODES);

  // --- layer 1: H1 = relu(agg(x @ W1) + b1)      (H1 ends in bufB)
  gemm_wmma_f32<NFEAT, NHID, 8><<<gemmBlk, nThreads, 0, stream>>>(x, W1, bufA, N_NODES);
  self_loop_init<128><<<nf128Blk, nThreads, 0, stream>>>(bufA, dinv, bufB, N_NODES);
  scatter128<<<edgeWaveBlk, nThreads, 0, stream>>>(bufA, src, dst, dinv, bufB, N_EDGES);
  bias_act<128, true><<<nf128Blk, nThreads, 0, stream>>>(bufB, b1, N_NODES);

  // --- layer 2: H2 = relu(agg(H1 @ Wm) + bm)     (H2 ends in bufB)
  gemm_wmma_f32<NHID, NHID, 8><<<gemmBlk, nThreads, 0, stream>>>(bufB, Wm, bufA, N_NODES);
  self_loop_init<128><<<nf128Blk, nThreads, 0, stream>>>(bufA, dinv, bufB, N_NODES);
  scatter128<<<edgeWaveBlk, nThreads, 0, stream>>>(bufA, src, dst, dinv, bufB, N_EDGES);
  bias_act<128, true><<<nf128Blk, nThreads, 0, stream>>>(bufB, bm, N_NODES);

  // --- layer 3: out = agg(H2 @ W2) + b2          (NCLASS=40, pad tiles to 48)
  gemm_wmma_f32<NHID, NCLASS, 3><<<gemmBlk, nThreads, 0, stream>>>(bufB, W2, bufA, N_NODES);
  self_loop_init<NCLASS><<<nf40Blk, nThreads, 0, stream>>>(bufA, dinv, out, N_NODES);
  scatterF<NCLASS><<<edgeWaveBlk, nThreads, 0, stream>>>(bufA, src, dst, dinv, out, N_EDGES);
  bias_act<NCLASS, false><<<nf40Blk, nThreads, 0, stream>>>(out, b2, N_NODES);
}